// GCNModel_61005715473011
// MI455X (gfx1250) — compile-verified
//
#include <hip/hip_runtime.h>
#include <math.h>

typedef __attribute__((ext_vector_type(2))) float v2f;
typedef __attribute__((ext_vector_type(8))) float v8f;

#define FD 64  // feature dim everywhere (F_IN == HID == EMB == 64)

// ---------------- utility kernels ----------------

__global__ void fill_f32(float* __restrict__ p, float v, int n) {
  int i = blockIdx.x * blockDim.x + threadIdx.x;
  if (i < n) p[i] = v;
}

__global__ void count_deg(const int* __restrict__ dst, float* __restrict__ deg, int E) {
  int e = blockIdx.x * blockDim.x + threadIdx.x;
  if (e < E) atomicAdd(&deg[dst[e]], 1.0f);
}

__global__ void deg_rsqrt(float* __restrict__ deg, int n) {
  int i = blockIdx.x * blockDim.x + threadIdx.x;
  if (i < n) deg[i] = rsqrtf(deg[i]);   // deg >= 1 always (self-loop)
}

// ---------------- fp32 WMMA GEMM: Y[N x 64] = X[N x 64] @ W[64 x 64] ----------------
// V_WMMA_F32_16X16X4_F32, wave32. One wave per 16-row block; 4 column tiles; K in steps of 4.
// A/B lane layout: element v of lane l holds k = v + 2*(l>=16), m/n = l%16.
// C/D: element v of lane l holds row m0 + v + 8*(l>=16), col n0 + l%16.
__global__ void gemm64_wmma(const float* __restrict__ X, const float* __restrict__ W,
                            float* __restrict__ Y, int nrows) {
  const int lane = threadIdx.x & 31;
  const int wave = threadIdx.x >> 5;
  const int m0 = (blockIdx.x * 8 + wave) * 16;
  if (m0 >= nrows) return;                 // uniform per wave
  const int lm = lane & 15;
  const int lh = lane >> 4;                // 0 or 1
  int row = m0 + lm;
  if (row >= nrows) row = nrows - 1;       // clamp loads; garbage rows not stored
  const float* __restrict__ xr = X + row * FD;

  v8f acc[4] = {};
#pragma unroll
  for (int k0 = 0; k0 < FD; k0 += 4) {
    const int ka = k0 + 2 * lh;
    v2f a;
    a.x = xr[ka];
    a.y = xr[ka + 1];
#pragma unroll
    for (int nt = 0; nt < 4; ++nt) {
      const int n0 = nt * 16;
      v2f b;
      b.x = W[ka * FD + n0 + lm];
      b.y = W[(ka + 1) * FD + n0 + lm];
      acc[nt] = __builtin_amdgcn_wmma_f32_16x16x4_f32(
          false, a, false, b, (short)0, acc[nt], false, false);
    }
  }
#pragma unroll
  for (int nt = 0; nt < 4; ++nt) {
    const int n0 = nt * 16;
#pragma unroll
    for (int v = 0; v < 8; ++v) {
      const int m = m0 + v + 8 * lh;
      if (m < nrows) Y[m * FD + n0 + lm] = acc[nt][v];
    }
  }
}

// ---------------- edge scatter: out[dst] += xw[src] * dis[src]*dis[dst] ----------------
// 64 consecutive threads handle one edge -> coalesced gather + coalesced f32 atomics.
// Edges e >= E are the self-loops (node i = e - E, norm = dis[i]^2).
__global__ void gcn_scatter(const float* __restrict__ xw, const int* __restrict__ src,
                            const int* __restrict__ dst, const float* __restrict__ dis,
                            float* __restrict__ out, int E, int N) {
  const int t = blockIdx.x * blockDim.x + threadIdx.x;
  const int total = (E + N) << 6;          // (1.1M)*64 = 70.4M < 2^31
  if (t >= total) return;
  const int f = t & 63;
  const int e = t >> 6;
  int s, d;
  float nrm;
  if (e < E) {
    s = src[e];
    d = dst[e];
    nrm = dis[s] * dis[d];
  } else {
    s = d = e - E;
    const float r = dis[s];
    nrm = r * r;
  }
  atomicAdd(&out[d * FD + f], xw[s * FD + f] * nrm);
}

__global__ void bias_act(float* __restrict__ buf, const float* __restrict__ b,
                         int n, int do_relu) {
  int i = blockIdx.x * blockDim.x + threadIdx.x;
  if (i >= n) return;
  float v = buf[i] + b[i & 63];
  buf[i] = do_relu ? fmaxf(v, 0.0f) : v;
}

// float atomic max via sign-split int/uint atomics (correct for mixed signs, -inf init)
__device__ __forceinline__ void atomicMaxF(float* addr, float val) {
  if (val >= 0.0f)
    atomicMax((int*)addr, __float_as_int(val));
  else
    atomicMin((unsigned int*)addr, __float_as_uint(val));
}

__global__ void seg_max(const float* __restrict__ ne, const int* __restrict__ batch,
                        float* __restrict__ ge, int N) {
  const int t = blockIdx.x * blockDim.x + threadIdx.x;
  if (t >= N * FD) return;
  const int f = t & 63;
  const int i = t >> 6;
  atomicMaxF(&ge[batch[i] * FD + f], ne[i * FD + f]);
}

__global__ void predict(const float* __restrict__ ge, const float* __restrict__ Wp,
                        const float* __restrict__ bp, float* __restrict__ pred, int G) {
  const int g = blockIdx.x * blockDim.x + threadIdx.x;
  if (g >= G) return;
  float s = bp[0];
#pragma unroll
  for (int f = 0; f < FD; ++f) s += ge[g * FD + f] * Wp[f];
  pred[g] = s;
}

// ---------------- launch ----------------

extern "C" void kernel_launch(void* const* d_in, const int* in_sizes, int n_in,
                              void* d_out, int out_size, void* d_ws, size_t ws_size,
                              hipStream_t stream) {
  const float* x    = (const float*)d_in[0];
  const int*   edge = (const int*)d_in[1];   // [2,E]: row0=src, row1=dst
  const int*   batch= (const int*)d_in[2];
  // d_in[3] = n_graphs (device scalar) -- G derived from sizes instead
  const float* W1 = (const float*)d_in[4];
  const float* b1 = (const float*)d_in[5];
  const float* W2 = (const float*)d_in[6];
  const float* b2 = (const float*)d_in[7];
  const float* Wp = (const float*)d_in[8];
  const float* bp = (const float*)d_in[9];

  const int N = in_sizes[0] / FD;
  const int E = in_sizes[1] / 2;
  const int G = (out_size - N * FD) / (FD + 1);  // pred[G] + ge[G*64] + ne[N*64]

  const int* src  = edge;
  const int* dstE = edge + E;

  float* pred = (float*)d_out;
  float* ge   = pred + G;
  float* ne   = ge + (size_t)G * FD;

  float* wsf = (float*)d_ws;
  float* dis = wsf;                        // N floats (deg -> rsqrt in place)
  float* xw  = wsf + N;                    // N*64 floats (xw, then h@W2)
  float* agg = xw + (size_t)N * FD;        // N*64 floats (layer-1 agg -> h)

  const int TB = 256;
  auto nb = [](long long n, int tb) { return (int)((n + tb - 1) / tb); };
  const int NF = N * FD;
  const int gemm_blocks = (((N + 15) / 16) + 7) / 8;

  // degree with self-loops: deg = 1 + count(dst); dis = rsqrt(deg)
  fill_f32<<<nb(N, TB), TB, 0, stream>>>(dis, 1.0f, N);
  count_deg<<<nb(E, TB), TB, 0, stream>>>(dstE, dis, E);
  deg_rsqrt<<<nb(N, TB), TB, 0, stream>>>(dis, N);

  // layer 1: h = relu(scatter(x@W1) + b1)
  fill_f32<<<nb(NF, TB), TB, 0, stream>>>(agg, 0.0f, NF);
  gemm64_wmma<<<gemm_blocks, 256, 0, stream>>>(x, W1, xw, N);
  gcn_scatter<<<nb((long long)(E + N) * FD, TB), TB, 0, stream>>>(xw, src, dstE, dis, agg, E, N);
  bias_act<<<nb(NF, TB), TB, 0, stream>>>(agg, b1, NF, 1);

  // layer 2: node_embed = scatter(h@W2) + b2   (h lives in agg, h@W2 reuses xw)
  gemm64_wmma<<<gemm_blocks, 256, 0, stream>>>(agg, W2, xw, N);
  fill_f32<<<nb(NF, TB), TB, 0, stream>>>(ne, 0.0f, NF);
  gcn_scatter<<<nb((long long)(E + N) * FD, TB), TB, 0, stream>>>(xw, src, dstE, dis, ne, E, N);
  bias_act<<<nb(NF, TB), TB, 0, stream>>>(ne, b2, NF, 0);

  // pool + head
  fill_f32<<<nb((long long)G * FD, TB), TB, 0, stream>>>(ge, -INFINITY, G * FD);
  seg_max<<<nb(NF, TB), TB, 0, stream>>>(ne, batch, ge, N);
  predict<<<nb(G, TB), TB, 0, stream>>>(ge, Wp, bp, pred, G);
}